// HaarDWT2D_9311489097777
// MI455X (gfx1250) — compile-verified
//
#include <hip/hip_runtime.h>

// Haar 2x2 DWT with per-channel band select (band = c % 4), 4 identical output
// copies. Pure streaming kernel: 453 MB in + 453 MB out @ 23.3 TB/s ~= 39 us
// floor on MI455X. No matrix math -> no WMMA. Optimizations:
//  - all loads/stores are b128 with non-temporal hints (footprint >> 192MB L2)
//  - grid-shaped index math lives in blockIdx/SGPRs, not per-lane VALU
//  - band = (n*96+c) & 3 == blockIdx.y & 3 (96 % 4 == 0) -> sign constants are
//    scalar-uniform, zero divides/modulos anywhere in the kernel

typedef __attribute__((ext_vector_type(4))) float v4f;

namespace {
constexpr int Nn  = 8;
constexpr int Cc  = 96;
constexpr int Hh  = 384;
constexpr int Ww  = 384;
constexpr int OHh = 192;
constexpr int OWw = 192;
constexpr int QUADS         = OWw / 4;        // 48 threads span one output row
constexpr int ROWS_PER_BLK  = 8;              // block = (48, 8) = 384 threads
constexpr int PLANE = Nn * Cc * OHh * OWw;    // 28,311,552 elems per output copy
}

__global__ __launch_bounds__(QUADS * ROWS_PER_BLK) void haar_dwt_band_select(
    const float* __restrict__ x, float* __restrict__ out) {
  const unsigned q    = threadIdx.x;                              // 0..47
  const unsigned oh   = blockIdx.x * ROWS_PER_BLK + threadIdx.y;  // 0..191
  const unsigned nc   = blockIdx.y;                               // n*96 + c, 0..767
  const unsigned band = nc & 3u;                                  // == c % 4

  // Input: 8 consecutive floats from each of two adjacent rows (2x b128 each).
  const float* base = x + ((size_t)nc * Hh + 2u * oh) * (size_t)Ww + 8u * q;
  v4f t0 = __builtin_nontemporal_load((const v4f*)base);            // a0 b0 a1 b1
  v4f t1 = __builtin_nontemporal_load((const v4f*)(base + 4));      // a2 b2 a3 b3
  v4f c0 = __builtin_nontemporal_load((const v4f*)(base + Ww));     // c0 d0 c1 d1
  v4f c1 = __builtin_nontemporal_load((const v4f*)(base + Ww + 4)); // c2 d2 c3 d3

  // band 0: +a+b+c+d | 1: -a-b+c+d | 2: -a+b-c+d | 3: +a-b-c+d  (all * 0.5)
  // Fold the 0.5 into the coefficients; all are wave/block-uniform scalars.
  const float fb = (band & 1u) ? -0.5f : 0.5f;   // coeff of b
  const float fc = (band & 2u) ? -0.5f : 0.5f;   // coeff of c
  const float fa = 2.0f * fb * fc;               // coeff of a (sign = sb*sc)
  const float fd = 0.5f;                          // coeff of d

  v4f r;
  r.x = fa * t0.x + fb * t0.y + fc * c0.x + fd * c0.y;
  r.y = fa * t0.z + fb * t0.w + fc * c0.z + fd * c0.w;
  r.z = fa * t1.x + fb * t1.y + fc * c1.x + fd * c1.y;
  r.w = fa * t1.z + fb * t1.w + fc * c1.z + fd * c1.w;

  // One b128 store per output copy (all four outputs are identical).
  const unsigned o = (nc * OHh + oh) * OWw + 4u * q;
  __builtin_nontemporal_store(r, (v4f*)(out + o));
  __builtin_nontemporal_store(r, (v4f*)(out + PLANE + o));
  __builtin_nontemporal_store(r, (v4f*)(out + 2 * PLANE + o));
  __builtin_nontemporal_store(r, (v4f*)(out + 3 * PLANE + o));
}

extern "C" void kernel_launch(void* const* d_in, const int* in_sizes, int n_in,
                              void* d_out, int out_size, void* d_ws, size_t ws_size,
                              hipStream_t stream) {
  (void)in_sizes; (void)n_in; (void)out_size; (void)d_ws; (void)ws_size;
  const float* x = (const float*)d_in[0];
  float* out = (float*)d_out;
  dim3 grid(OHh / ROWS_PER_BLK, Nn * Cc);   // (24, 768)
  dim3 block(QUADS, ROWS_PER_BLK);          // (48, 8) = 384 threads, 12 waves
  haar_dwt_band_select<<<grid, block, 0, stream>>>(x, out);
}